// ConvCapsLayer_41154376630983
// MI455X (gfx1250) — compile-verified
//
#include <hip/hip_runtime.h>

// ---------------------------------------------------------------------------
// ConvCaps layer + dynamic routing, fused in LDS. fp32 WMMA (16x16x4, K=8 via
// two chained WMMAs). One workgroup = 8 (b,l) pairs; 8 wave32 waves; wave w
// exclusively owns output caps n = {w, w+8, w+16, w+24}.
// Routing is fully peeled/specialized:
//   s0 (uniform c=1/32, pure WMMA C-chain) -> squash -> db0 -> softmax
//   s1 (c folded into A rows)              -> squash -> db1 -> softmax
//   s2                                     -> squash -> out
// ---------------------------------------------------------------------------

typedef float    v2f __attribute__((ext_vector_type(2)));
typedef float    v8f __attribute__((ext_vector_type(8)));
typedef _Float16 v8h __attribute__((ext_vector_type(8)));

#define J_TOT   288
#define LBLK    8
#define L_TOT   196
#define NBLK    98          // 784 / LBLK

// LDS layout (bytes):
//   p    [288][8][8]  f32   73728   patches, [j][l][d]
//   blog [288][8][32] f16  147456   routing logits, [j][l][n] (contiguous in n)
//   m_l  [288][8]     f32    9216   per-(j,l) logit row max
//   rz   [288][8]     f32    9216   per-(j,l) 1/sum(exp)
//   vl   [8][32][16]  f32   16384   s / v vectors, [l][n][dout]
#define OFF_P    0
#define OFF_B    73728
#define OFF_M    (73728 + 147456)
#define OFF_RZ   (OFF_M + 9216)
#define OFF_V    (OFF_RZ + 9216)
#define SMEM_SZ  (OFF_V + 16384)     // 256000 B < 320 KB

// ---- s-pass: s[l,n,:] = sum_j c[l,j,n] * u_hat[l,j,n,:]  (A = p, B = W) ----
template<bool UNIFORM>
__device__ __forceinline__
void s_pass(const float* __restrict__ W, const float* __restrict__ p,
            const _Float16* __restrict__ blog, const float* __restrict__ m_l,
            const float* __restrict__ rz, float* __restrict__ vl,
            int wave, int lane) {
    const int ln16  = lane & 15;
    const int halfK = lane >> 4;
    const int lrow  = lane & 7;
    const v8f zero8 = {0.f,0.f,0.f,0.f,0.f,0.f,0.f,0.f};
    v8f acc[4] = {zero8, zero8, zero8, zero8};

    for (int j = 0; j < J_TOT; ++j) {
        // A operand: rows = l (lane ln16), K = d; vgpr0 holds K=2h, vgpr1 K=2h+1
        v2f alo = {0.f, 0.f}, ahi = {0.f, 0.f};
        if (ln16 < LBLK) {
            const float* pb = &p[j * 64 + ln16 * 8 + 2 * halfK];
            alo = *(const v2f*)pb;
            ahi = *(const v2f*)(pb + 4);
        }
        float cv = 0.0f;
        if (!UNIFORM) {
            // lane s*8+r holds c for (n = wave + s*8, l = r): one exp per lane
            int nc = wave + ((lane >> 3) << 3);
            cv = __expf((float)blog[j * 256 + lrow * 32 + nc] - m_l[j * 8 + lrow])
                 * rz[j * 8 + lrow];
        }
#pragma unroll
        for (int ni = 0; ni < 4; ++ni) {
            const int n = wave + ni * 8;
            const float* wb = &W[(size_t)(j * 512 + n * 16 + ln16) * 8 + 2 * halfK];
            const v2f blo = *(const v2f*)wb;
            const v2f bhi = *(const v2f*)(wb + 4);
            v2f ua = alo, uh = ahi;
            if (!UNIFORM) {
                // c is a per-lane row scalar -> fold into A instead of weighting D
                const float c = __shfl(cv, (ni << 3) | lrow, 32);
                ua = alo * c;
                uh = ahi * c;
            }
            acc[ni] = __builtin_amdgcn_wmma_f32_16x16x4_f32(
                          false, uh, false, bhi, (short)0, acc[ni], false, false);
            acc[ni] = __builtin_amdgcn_wmma_f32_16x16x4_f32(
                          false, ua, false, blo, (short)0, acc[ni], false, false);
        }
    }
    if (halfK == 0) {       // lanes 0-15 hold rows l=0..7 (valid block rows)
        const float sc = UNIFORM ? 0.03125f : 1.0f;
#pragma unroll
        for (int ni = 0; ni < 4; ++ni) {
            const int n = wave + ni * 8;
#pragma unroll
            for (int k = 0; k < 8; ++k)
                vl[(k * 32 + n) * 16 + ln16] = acc[ni][k] * sc;
        }
    }
}

// ---- logit update: b[l,j,n] += <v[l,n,:], u_hat[l,j,n,:]> -----------------
// Transposed WMMA (A = W rows=dout, B = p cols=l): identical loads, and the
// dout reduction becomes an in-lane 8-FMA + one shfl_xor(16).
template<bool FIRST>
__device__ __forceinline__
void b_pass(const float* __restrict__ W, const float* __restrict__ p,
            _Float16* __restrict__ blog, const float* __restrict__ vl,
            int wave, int lane) {
    const int ln16  = lane & 15;
    const int halfK = lane >> 4;
    const int lcl   = lane & 7;
    const v8f zero8 = {0.f,0.f,0.f,0.f,0.f,0.f,0.f,0.f};

    for (int j = 0; j < J_TOT; ++j) {
        v2f blo = {0.f, 0.f}, bhi = {0.f, 0.f};     // p as B (cols = l)
        if (ln16 < LBLK) {
            const float* pb = &p[j * 64 + ln16 * 8 + 2 * halfK];
            blo = *(const v2f*)pb;
            bhi = *(const v2f*)(pb + 4);
        }
#pragma unroll
        for (int ni = 0; ni < 4; ++ni) {
            const int n = wave + ni * 8;
            const float* wb = &W[(size_t)(j * 512 + n * 16 + ln16) * 8 + 2 * halfK];
            const v2f alo = *(const v2f*)wb;        // W as A (rows = dout)
            const v2f ahi = *(const v2f*)(wb + 4);
            v8f U = __builtin_amdgcn_wmma_f32_16x16x4_f32(
                        false, ahi, false, bhi, (short)0, zero8, false, false);
            U = __builtin_amdgcn_wmma_f32_16x16x4_f32(
                        false, alo, false, blo, (short)0, U, false, false);
            // U[k] = u_hat[l=ln16, j, n, dout = k + 8*halfK]; cols ln16>=8 are 0
            const float* vv = &vl[(lcl * 32 + n) * 16 + 8 * halfK];
            const float4 v0 = *(const float4*)vv;
            const float4 v1 = *(const float4*)(vv + 4);
            float t = U[0]*v0.x + U[1]*v0.y + U[2]*v0.z + U[3]*v0.w
                    + U[4]*v1.x + U[5]*v1.y + U[6]*v1.z + U[7]*v1.w;
            t += __shfl_xor(t, 16, 32);             // combine dout halves
            if (lane < 8) {
                _Float16* bp = &blog[j * 256 + lane * 32 + n];
                const float ob = FIRST ? 0.0f : (float)(*bp);
                *bp = (_Float16)(ob + t);
            }
        }
    }
}

__global__ __launch_bounds__(256)
void caps_routing_wmma(const float* __restrict__ u,
                       const float* __restrict__ W,
                       float* __restrict__ out) {
    extern __shared__ char smem[];
    float*    p    = (float*)(smem + OFF_P);
    _Float16* blog = (_Float16*)(smem + OFF_B);
    float*    m_l  = (float*)(smem + OFF_M);
    float*    rz   = (float*)(smem + OFF_RZ);
    float*    vl   = (float*)(smem + OFF_V);

    const int tid = threadIdx.x;
    const int bl0 = blockIdx.x * LBLK;

    // ---- gather patches p[j][l][d] (faithful scrambled unfold view) ----
    for (int t = tid; t < J_TOT * LBLK * 8; t += 256) {
        int j   = t >> 6;
        int rem = t & 63;
        int lr  = rem >> 3;
        int d   = rem & 7;
        int bl  = bl0 + lr;
        int b   = bl / L_TOT;
        int l   = bl - b * L_TOT;
        int flat = l * (J_TOT * 8) + j * 8 + d;
        int ch   = flat / L_TOT;
        int lpos = flat - ch * L_TOT;
        int cc   = ch / 9;
        int off  = ch - cc * 9;
        int ki   = off / 3;
        int kj   = off - ki * 3;
        int hy   = lpos / 14;
        int wx   = lpos - hy * 14;
        int h    = hy + ki - 1;
        int w    = wx + kj - 1;
        float val = 0.0f;
        if ((unsigned)h < 14u && (unsigned)w < 14u)
            val = u[(b * 256 + cc) * L_TOT + h * 14 + w];
        p[j * 64 + lr * 8 + d] = val;
    }
    __syncthreads();

    const int wave = tid >> 5;
    const int lane = tid & 31;

    auto squash = [&]() {
        const int lr = tid >> 5;
        const int n  = tid & 31;
        float* sp = &vl[(lr * 32 + n) * 16];
        float4 s0 = *(float4*)(sp + 0),  s1 = *(float4*)(sp + 4);
        float4 s2 = *(float4*)(sp + 8),  s3 = *(float4*)(sp + 12);
        float sq = s0.x*s0.x + s0.y*s0.y + s0.z*s0.z + s0.w*s0.w
                 + s1.x*s1.x + s1.y*s1.y + s1.z*s1.z + s1.w*s1.w
                 + s2.x*s2.x + s2.y*s2.y + s2.z*s2.z + s2.w*s2.w
                 + s3.x*s3.x + s3.y*s3.y + s3.z*s3.z + s3.w*s3.w;
        const float scale = sq / ((1.0f + sq) * sqrtf(sq));
        s0.x*=scale; s0.y*=scale; s0.z*=scale; s0.w*=scale;
        s1.x*=scale; s1.y*=scale; s1.z*=scale; s1.w*=scale;
        s2.x*=scale; s2.y*=scale; s2.z*=scale; s2.w*=scale;
        s3.x*=scale; s3.y*=scale; s3.z*=scale; s3.w*=scale;
        *(float4*)(sp + 0) = s0;  *(float4*)(sp + 4)  = s1;
        *(float4*)(sp + 8) = s2;  *(float4*)(sp + 12) = s3;
    };

    auto softmax_norm = [&]() {
        for (int idx = tid; idx < J_TOT * LBLK; idx += 256) {
            const int j  = idx >> 3;
            const int lr = idx & 7;
            const v8h* bp = (const v8h*)&blog[j * 256 + lr * 32]; // 64B row
            const v8h q0 = bp[0], q1 = bp[1], q2 = bp[2], q3 = bp[3];
            float f[32];
#pragma unroll
            for (int k = 0; k < 8; ++k) {
                f[k]      = (float)q0[k];
                f[k + 8]  = (float)q1[k];
                f[k + 16] = (float)q2[k];
                f[k + 24] = (float)q3[k];
            }
            float mx = f[0];
#pragma unroll
            for (int k = 1; k < 32; ++k) mx = fmaxf(mx, f[k]);
            float s = 0.0f;
#pragma unroll
            for (int k = 0; k < 32; ++k) s += __expf(f[k] - mx);
            m_l[j * 8 + lr] = mx;
            rz[j * 8 + lr]  = 1.0f / s;
        }
    };

    // ---- iteration 0: c = softmax(0) = 1/32 ----
    s_pass<true>(W, p, blog, m_l, rz, vl, wave, lane);
    __syncthreads();
    squash();
    __syncthreads();
    b_pass<true>(W, p, blog, vl, wave, lane);
    __syncthreads();
    softmax_norm();
    __syncthreads();

    // ---- iteration 1 ----
    s_pass<false>(W, p, blog, m_l, rz, vl, wave, lane);
    __syncthreads();
    squash();
    __syncthreads();
    b_pass<false>(W, p, blog, vl, wave, lane);
    __syncthreads();
    softmax_norm();
    __syncthreads();

    // ---- iteration 2 (final: no logit update) ----
    s_pass<false>(W, p, blog, m_l, rz, vl, wave, lane);
    __syncthreads();
    squash();
    __syncthreads();

    // ---- emit v -> out[(bl*32 + n)*16 + dout] ----
    {
        const int lr = tid >> 5;
        const int n  = tid & 31;
        const int bl = bl0 + lr;
        const float4* sp = (const float4*)&vl[(lr * 32 + n) * 16];
        float4* op = (float4*)&out[(size_t)(bl * 32 + n) * 16];
        op[0] = sp[0]; op[1] = sp[1]; op[2] = sp[2]; op[3] = sp[3];
    }
}

extern "C" void kernel_launch(void* const* d_in, const int* in_sizes, int n_in,
                              void* d_out, int out_size, void* d_ws, size_t ws_size,
                              hipStream_t stream) {
    (void)in_sizes; (void)n_in; (void)out_size; (void)d_ws; (void)ws_size;
    const float* u = (const float*)d_in[0];   // (4,32,8,196) f32
    const float* W = (const float*)d_in[1];   // (288,512,8)  f32
    float* out = (float*)d_out;               // (4,6272,16)  f32

    (void)hipFuncSetAttribute((const void*)caps_routing_wmma,
                              hipFuncAttributeMaxDynamicSharedMemorySize,
                              (int)SMEM_SZ);
    caps_routing_wmma<<<NBLK, 256, SMEM_SZ, stream>>>(u, W, out);
}